// ConcatAttention_12652973654064
// MI455X (gfx1250) — compile-verified
//
#include <hip/hip_runtime.h>
#include <hip/hip_bf16.h>

#define HS 128   // hidden
#define BB 1024  // batch
#define SS 512   // seq

typedef __attribute__((ext_vector_type(2))) float v2f;
typedef __attribute__((ext_vector_type(8))) float v8f;

// ---------------------------------------------------------------------------
// Kernel 1: v[b,h] = sum_k fh[b,k] * W[k,h]   (fp32 WMMA 16x16x4, 1 wave/tile)
// ---------------------------------------------------------------------------
__global__ __launch_bounds__(256) void kv_gemm(const float* __restrict__ fh,
                                               const float* __restrict__ W,
                                               float* __restrict__ v) {
  int wave = (blockIdx.x * 256 + threadIdx.x) >> 5;   // 512 tiles total
  int lane = threadIdx.x & 31;
  int m0 = (wave >> 3) * 16;     // 64 M-tiles over B
  int n0 = (wave & 7) * 16;      // 8  N-tiles over H
  int half = lane >> 4;          // lane group: K offset +2
  int l = lane & 15;
  v8f c = {};
  #pragma unroll
  for (int k = 0; k < HS; k += 4) {
    int ka = k + half * 2;
    v2f a, bf;
    a.x  = fh[(m0 + l) * HS + ka];
    a.y  = fh[(m0 + l) * HS + ka + 1];
    bf.x = W[(ka) * HS + n0 + l];
    bf.y = W[(ka + 1) * HS + n0 + l];
    c = __builtin_amdgcn_wmma_f32_16x16x4_f32(false, a, false, bf,
                                              (short)0, c, false, false);
  }
  #pragma unroll
  for (int r = 0; r < 8; ++r)
    v[(m0 + r + half * 8) * HS + n0 + l] = c[r];
}

// ---------------------------------------------------------------------------
// Kernel 2: per batch row b (one WGP each):
//   TDM-load X[b] (512x128 f32, 256KB) -> LDS with row pad (stride 130 f32),
//   scores[s] = X[b,s,:].v[b]  ->  softmax over s  ->  cntxt[h] = sum_s X w
// ---------------------------------------------------------------------------
__global__ __launch_bounds__(512) void attn_fused(const float* __restrict__ X,
                                                  const float* __restrict__ v,
                                                  float* __restrict__ wout,
                                                  float* __restrict__ cout) {
  extern __shared__ float lds[];
  const int tid = threadIdx.x;
  const int b = blockIdx.x;
  const int RS = HS + 2;                 // padded LDS row stride (bank-safe)
  float* Xl   = lds;                     // 512*130 floats
  float* vl   = Xl + SS * RS;            // 128
  float* wl   = vl + HS;                 // 512
  float* red  = wl + SS;                 // 32
  float* part = red + 32;                // 512

  // ---- stage X[b] into LDS (single pass over HBM) ----
#if __has_builtin(__builtin_amdgcn_tensor_load_to_lds)
  {
    typedef __attribute__((ext_vector_type(4))) unsigned int u32x4;
    typedef __attribute__((ext_vector_type(8))) int i32x8;
    typedef __attribute__((ext_vector_type(4))) int i32x4;
    unsigned long long ga =
        (unsigned long long)(const void*)(X + (size_t)b * SS * HS);
    unsigned int lds_off = (unsigned int)(unsigned long long)(void*)Xl;
    // D# group0: count=1 | lds_addr | global_addr[56:0] | type=2
    u32x4 g0 = {1u, lds_off, (unsigned int)ga,
                ((unsigned int)((ga >> 32) & 0x01FFFFFFu)) | 0x80000000u};
    // D# group1: data_size=4B, pad_enable, pad_interval=128dw, pad_amount=2dw,
    //            tensor_dim0=128, tensor_dim1=512, tile 128x512, stride0=128
    i32x8 g1 = {0x03920000, 0x00800000, 0x02000000, 0x00800000,
                0x00000200, 128, 0, 0};
    i32x4 gz4 = {0, 0, 0, 0};
    i32x8 gz8 = {0, 0, 0, 0, 0, 0, 0, 0};
    if (tid == 0) {
      __builtin_amdgcn_tensor_load_to_lds(g0, g1, gz4, gz4, gz8, 0);
      __builtin_amdgcn_s_wait_tensorcnt(0);
    }
  }
#else
  for (int i = tid; i < SS * HS / 2; i += 512) {
    int s = i >> 6, h2 = (i & 63) << 1;
    *(float2*)&Xl[s * RS + h2] =
        *(const float2*)&X[(size_t)b * SS * HS + s * HS + h2];
  }
#endif
  if (tid < HS) vl[tid] = v[(size_t)b * HS + tid];
  __syncthreads();

  // ---- scores: thread t owns s = t (LDS reads conflict-free via pad) ----
  float sc = 0.f;
  {
    const float2* xr = (const float2*)&Xl[tid * RS];
    const float2* vr = (const float2*)vl;
    #pragma unroll 8
    for (int i = 0; i < HS / 2; ++i) {
      float2 xv = xr[i], vv = vr[i];
      sc += xv.x * vv.x + xv.y * vv.y;
    }
  }

  // ---- softmax over s (wave32 shuffle + 16-wave LDS reduction) ----
  float mx = sc;
  for (int m = 16; m; m >>= 1) mx = fmaxf(mx, __shfl_xor(mx, m, 32));
  if ((tid & 31) == 0) red[tid >> 5] = mx;
  __syncthreads();
  if (tid < 32) {
    float y = (tid < 16) ? red[tid] : -3.4e38f;
    for (int m = 8; m; m >>= 1) y = fmaxf(y, __shfl_xor(y, m, 32));
    if (tid == 0) red[16] = y;
  }
  __syncthreads();
  mx = red[16];
  float e = __expf(sc - mx);
  float sm = e;
  for (int m = 16; m; m >>= 1) sm += __shfl_xor(sm, m, 32);
  __syncthreads();
  if ((tid & 31) == 0) red[tid >> 5] = sm;
  __syncthreads();
  if (tid < 32) {
    float y = (tid < 16) ? red[tid] : 0.f;
    for (int m = 8; m; m >>= 1) y += __shfl_xor(y, m, 32);
    if (tid == 0) red[16] = y;
  }
  __syncthreads();
  float wgt = e * (1.0f / red[16]);
  wl[tid] = wgt;
  wout[(size_t)b * SS + tid] = wgt;
  __syncthreads();

  // ---- context: cntxt[h] = sum_s X[s,h]*w[s]; 4 partials per h ----
  {
    int h = tid & (HS - 1);
    int q = tid >> 7;                        // 0..3
    float acc = 0.f;
    int s0 = q * (SS / 4);
    for (int s = s0; s < s0 + SS / 4; ++s)
      acc += Xl[s * RS + h] * wl[s];
    part[q * HS + h] = acc;
  }
  __syncthreads();
  if (tid < HS) {
    float cx = part[tid] + part[HS + tid] + part[2 * HS + tid] +
               part[3 * HS + tid];
    cout[(size_t)b * HS + tid] = cx;
  }
}

// ---------------------------------------------------------------------------
// Kernel 3: att_hidden = tanh([cntxt, fh] @ align_w^T + align_b)
//           M=1024, N=128, K=256  (fp32 WMMA 16x16x4)
// ---------------------------------------------------------------------------
__global__ __launch_bounds__(256) void final_gemm(const float* __restrict__ cn,
                                                  const float* __restrict__ fh,
                                                  const float* __restrict__ aw,
                                                  const float* __restrict__ ab,
                                                  float* __restrict__ out) {
  int wave = (blockIdx.x * 256 + threadIdx.x) >> 5;
  int lane = threadIdx.x & 31;
  int m0 = (wave >> 3) * 16;
  int n0 = (wave & 7) * 16;
  int half = lane >> 4;
  int l = lane & 15;
  v8f c = {};
  #pragma unroll
  for (int k = 0; k < 2 * HS; k += 4) {
    int ka = k + half * 2;
    const float* A = (ka < HS) ? cn : fh;   // concat without materializing
    int col = ka & (HS - 1);
    v2f a, bf;
    a.x  = A[(m0 + l) * HS + col];
    a.y  = A[(m0 + l) * HS + col + 1];
    bf.x = aw[(n0 + l) * (2 * HS) + ka];     // B[k,n] = align_w[n, k]
    bf.y = aw[(n0 + l) * (2 * HS) + ka + 1];
    c = __builtin_amdgcn_wmma_f32_16x16x4_f32(false, a, false, bf,
                                              (short)0, c, false, false);
  }
  float bias = ab[n0 + l];
  #pragma unroll
  for (int r = 0; r < 8; ++r)
    out[(m0 + r + half * 8) * HS + n0 + l] = tanhf(c[r] + bias);
}

// ---------------------------------------------------------------------------
extern "C" void kernel_launch(void* const* d_in, const int* in_sizes, int n_in,
                              void* d_out, int out_size, void* d_ws,
                              size_t ws_size, hipStream_t stream) {
  const float* X  = (const float*)d_in[0];  // [B,S,H]
  const float* fh = (const float*)d_in[1];  // [B,H]
  const float* Wl = (const float*)d_in[2];  // [H,H]
  // d_in[3] (linear_b) is provably unused: constant-per-row shift cancels in softmax
  const float* aw = (const float*)d_in[4];  // [H,2H]
  const float* ab = (const float*)d_in[5];  // [H]

  float* out_hidden  = (float*)d_out;               // [B,H]
  float* out_weights = out_hidden + BB * HS;        // [B,S]
  float* v  = (float*)d_ws;                         // [B,H]
  float* cn = v + BB * HS;                          // [B,H]

  kv_gemm<<<64, 256, 0, stream>>>(fh, Wl, v);

  const int lds_bytes = (SS * (HS + 2) + HS + SS + 32 + SS) * 4;  // 270,976 B
  attn_fused<<<BB, 512, lds_bytes, stream>>>(X, v, out_weights, cn);

  final_gemm<<<64, 256, 0, stream>>>(cn, fh, aw, ab, out_hidden);
}